// MPLayer_7928509628988
// MI455X (gfx1250) — compile-verified
//
#include <hip/hip_runtime.h>

#define D      128
#define NNODES 50000
#define NEDGES 600000

typedef __attribute__((ext_vector_type(2))) float v2f;
typedef __attribute__((ext_vector_type(4))) float v4f;
typedef __attribute__((ext_vector_type(8))) float v8f;

#if defined(__has_builtin)
#if __has_builtin(__builtin_amdgcn_wmma_f32_16x16x4_f32)
#define HAVE_WMMA_F32 1
#endif
#endif
#ifndef HAVE_WMMA_F32
#define HAVE_WMMA_F32 0
#endif

// ---------------------------------------------------------------------------
// 128x128 transpose: Wt[n][k] = W[k][n]. Tiny (64 KB), one-shot.
// ---------------------------------------------------------------------------
__global__ __launch_bounds__(256) void transpose128(const float* __restrict__ W,
                                                    float* __restrict__ Wt) {
  const int i = blockIdx.x * 256 + threadIdx.x;  // 0..16383
  const int k = i >> 7;
  const int n = i & 127;
  Wt[(size_t)n * D + k] = W[i];
}

// ---------------------------------------------------------------------------
// GEMM: out[M x 128] = act( X[M x 128] @ W[128 x 128] (+ bias) ), W given
// TRANSPOSED (Wt[n][k]) so both fragments are contiguous b64 loads.
// One wave32 computes one 16x16 output tile via V_WMMA_F32_16X16X4_F32.
// Block = 256 threads = 8 waves -> one 16-row strip x all 8 column tiles.
//
// A-fragment (32-bit A 16x4, ISA 7.12.2):
//   lanes 0-15 : M = lane,    VGPR0/1 = K k0+0,k0+1
//   lanes 16-31: M = lane-16, VGPR0/1 = K k0+2,k0+3
// B-fragment mirrored on N. C/D: VGPR r -> M = r + 8*(lane>=16), N = lane&15.
// ---------------------------------------------------------------------------
template <bool FUSE_BIAS_RELU>
__global__ __launch_bounds__(256) void gemm128_wmma(const float* __restrict__ X,
                                                    const float* __restrict__ Wt,
                                                    const float* __restrict__ bias,
                                                    float* __restrict__ out) {
  const int wave = threadIdx.x >> 5;  // 0..7 -> column tile
  const int lane = threadIdx.x & 31;
  const int half = lane >> 4;
  const int l16  = lane & 15;
  const int m0   = blockIdx.x * 16;
  const int col  = wave * 16 + l16;

  v8f acc = {0.f, 0.f, 0.f, 0.f, 0.f, 0.f, 0.f, 0.f};

#if HAVE_WMMA_F32
  // Per-lane K base offset: lanes 0-15 take K=k0,k0+1; lanes 16-31 K=k0+2,k0+3.
  const float* xrow = X + (size_t)(m0 + l16) * D + 2 * half;
  const float* wrow = Wt + (size_t)col * D + 2 * half;
#pragma unroll
  for (int k0 = 0; k0 < D; k0 += 4) {
    v2f a = *(const v2f*)(xrow + k0);  // one global_load_b64
    v2f b = *(const v2f*)(wrow + k0);  // one global_load_b64
    acc = __builtin_amdgcn_wmma_f32_16x16x4_f32(
        /*neg_a=*/false, a, /*neg_b=*/false, b,
        /*c_mod=*/(short)0, acc, /*reuse_a=*/false, /*reuse_b=*/false);
  }
#else
  // Scalar fallback producing the identical D-matrix lane mapping.
#pragma unroll
  for (int r = 0; r < 8; ++r) {
    const int m = m0 + r + 8 * half;
    float s = 0.f;
    for (int k = 0; k < D; ++k)
      s = fmaf(X[(size_t)m * D + k], Wt[(size_t)col * D + k], s);
    acc[r] = s;
  }
#endif

#pragma unroll
  for (int r = 0; r < 8; ++r) {
    const int m = m0 + r + 8 * half;
    float v = acc[r];
    if (FUSE_BIAS_RELU) {
      v += bias[col];
      v = v > 0.f ? v : 0.f;
    }
    out[(size_t)m * D + col] = v;
  }
}

// ---------------------------------------------------------------------------
// Zero the aggregation buffer (float4 stores).
// ---------------------------------------------------------------------------
__global__ __launch_bounds__(256) void zero_f4(v4f* __restrict__ p, int n4) {
  int i = blockIdx.x * 256 + threadIdx.x;
  if (i < n4) p[i] = (v4f){0.f, 0.f, 0.f, 0.f};
}

// ---------------------------------------------------------------------------
// Per-edge: msg = relu(edge_w * Y[src] + b_msg); agg[dst] += msg.
// One wave32 per edge, 4 features per lane, L2-resident f32 atomics.
// ---------------------------------------------------------------------------
__global__ __launch_bounds__(256) void edge_scatter(const float* __restrict__ Y,
                                                    const float* __restrict__ ef,
                                                    const int* __restrict__ src,
                                                    const int* __restrict__ dst,
                                                    const float* __restrict__ bmsg,
                                                    float* __restrict__ agg, int E) {
  const int t = blockIdx.x * 256 + threadIdx.x;
  const int e = t >> 5;
  if (e >= E) return;
  const int base = (t & 31) * 4;

  const int   s = src[e];
  const int   d = dst[e];
  const float w = ef[e];

  v4f y = *(const v4f*)(Y + (size_t)s * D + base);
  v4f b = *(const v4f*)(bmsg + base);

  float m0 = fmaxf(fmaf(w, y.x, b.x), 0.f);
  float m1 = fmaxf(fmaf(w, y.y, b.y), 0.f);
  float m2 = fmaxf(fmaf(w, y.z, b.z), 0.f);
  float m3 = fmaxf(fmaf(w, y.w, b.w), 0.f);

  float* ap = agg + (size_t)d * D + base;
  atomicAdd(ap + 0, m0);
  atomicAdd(ap + 1, m1);
  atomicAdd(ap + 2, m2);
  atomicAdd(ap + 3, m3);
}

extern "C" void kernel_launch(void* const* d_in, const int* in_sizes, int n_in,
                              void* d_out, int out_size, void* d_ws, size_t ws_size,
                              hipStream_t stream) {
  const float* node_feats = (const float*)d_in[0];
  const float* edge_feats = (const float*)d_in[1];
  const int*   src        = (const int*)d_in[2];
  const int*   dst        = (const int*)d_in[3];
  const float* W_msg      = (const float*)d_in[4];
  const float* b_msg      = (const float*)d_in[5];
  const float* W_out      = (const float*)d_in[6];
  const float* b_out      = (const float*)d_in[7];
  float*       out        = (float*)d_out;

  float* Y      = (float*)d_ws;                  // [NNODES,128] 25.6 MB
  float* agg    = Y + (size_t)NNODES * D;        // [NNODES,128] 25.6 MB
  float* Wt_msg = agg + (size_t)NNODES * D;      // [128,128]    64 KB
  float* Wt_out = Wt_msg + D * D;                // [128,128]    64 KB

  // 0) transpose both weight matrices for b64 B-fragment loads
  transpose128<<<D * D / 256, 256, 0, stream>>>(W_msg, Wt_msg);
  transpose128<<<D * D / 256, 256, 0, stream>>>(W_out, Wt_out);

  // 1) Y = node_feats @ W_msg  (per-edge scalar commutes with the matmul)
  gemm128_wmma<false><<<NNODES / 16, 256, 0, stream>>>(node_feats, Wt_msg, nullptr, Y);

  // 2) agg = 0
  const int n4 = NNODES * D / 4;
  zero_f4<<<(n4 + 255) / 256, 256, 0, stream>>>((v4f*)agg, n4);

  // 3) agg[dst] += relu(w_e * Y[src] + b_msg)   (L2-resident scatter)
  edge_scatter<<<(NEDGES * 32 + 255) / 256, 256, 0, stream>>>(Y, edge_feats, src, dst,
                                                              b_msg, agg, NEDGES);

  // 4) out = relu(agg @ W_out + b_out)
  gemm128_wmma<true><<<NNODES / 16, 256, 0, stream>>>(agg, Wt_out, b_out, out);
}